// RWKV7Attention_42125039239451
// MI455X (gfx1250) — compile-verified
//
#include <hip/hip_runtime.h>
#include <hip/hip_bf16.h>
#include <math.h>

// ---------------------------------------------------------------------------
// RWKV7 attention forward for MI455X (gfx1250, wave32).
//  - fp32 WMMA (v_wmma_f32_16x16x4_f32) for all dense / LoRA GEMMs
//    (64x256 block tile, 32x64 wave tile, BK=32, pair-interleaved B in LDS
//     so every fragment is one aligned ds_load_b64, bank-conflict-free)
//  - per-(b,h) recurrent scan kernel, one state row per lane, with
//    double-buffered global loads hidden under the state-update FMAs
//  - token-shift mixing fused into GEMM A-tile loads
// ---------------------------------------------------------------------------

typedef __attribute__((ext_vector_type(2))) float v2f;
typedef __attribute__((ext_vector_type(8))) float v8f;

#define D_MODEL 1024
#define NHEADS  16
#define HEADD   64
#define TLEN    2048
#define DECAY_SCALE (-0.6065306597126334f)
#define GN_EPS  (64.0f * 1e-5f)

#define BM 64
#define BN 256
#define BK 32
#define APAD 34            // As row stride (even -> aligned b64, conflict-free)
#define BPADC (BN + 16)    // B pair-row stride in float2 (pair-row delta = 32 banks)

// act: 0 = none, 1 = tanh, 2 = sigmoid, 3 = DECAY_SCALE * sigmoid
// mix != nullptr: A row m is the token-shift-mixed hidden state:
//   xmix[m][c] = x[m][c]*(1-mix[c]) + x[m-1][c]*mix[c]   (x[-1] = 0 at t==0)
__global__ __launch_bounds__(256) void wmma_gemm_f32(
    const float* __restrict__ A, const float* __restrict__ mix,
    const float* __restrict__ W, const float* __restrict__ bias,
    float* __restrict__ Y, int M, int N, int K, int act)
{
    __shared__ float As[BM][APAD];          // 64 x 32 tile, stride 34
    __shared__ float Bs[BK / 2][BPADC * 2]; // K-pair interleaved: [p][col*2 + (k&1)]

    const int tid  = threadIdx.x;
    const int lane = tid & 31;              // wave32
    const int wave = tid >> 5;              // 0..7
    const int wm   = wave >> 2;             // 0..1  (row sub-tile, 32 rows)
    const int wn   = wave & 3;              // 0..3  (col sub-tile, 64 cols)
    const int L2   = lane & 15;
    const int kh   = (lane >> 4) << 1;      // 0 or 2 : K-half per ISA A/B layout

    v8f acc[8];
#pragma unroll
    for (int q = 0; q < 8; ++q)
#pragma unroll
        for (int r = 0; r < 8; ++r) acc[q][r] = 0.0f;

    const int m0 = blockIdx.y * BM;
    const int n0 = blockIdx.x * BN;

    for (int kc = 0; kc < K; kc += BK) {
        // ---- stage A tile (64 x 32) ----
        for (int i = tid; i < BM * BK; i += 256) {
            int row = i / BK, kk = i % BK;
            int m = m0 + row;
            int kg = kc + kk;
            float val;
            if (mix) {
                float xv = A[(size_t)m * K + kg];
                int t = m % TLEN;
                float prev = (t == 0) ? 0.0f : A[(size_t)(m - 1) * K + kg];
                val = xv + (prev - xv) * mix[kg];
            } else {
                val = A[(size_t)m * K + kg];
            }
            As[row][kk] = val;
        }
        // ---- stage B tile (32 x 256), K-pair interleaved ----
        for (int i = tid; i < BK * BN; i += 256) {
            int kk = i / BN, col = i % BN;
            int n = n0 + col;
            float val = (n < N) ? W[(size_t)(kc + kk) * N + n] : 0.0f;
            Bs[kk >> 1][col * 2 + (kk & 1)] = val;
        }
        __syncthreads();

#pragma unroll
        for (int k0 = 0; k0 < BK; k0 += 4) {
            v2f a[2], b[4];
#pragma unroll
            for (int s = 0; s < 2; ++s)
                a[s] = *(const v2f*)&As[wm * 32 + s * 16 + L2][k0 + kh];
#pragma unroll
            for (int n = 0; n < 4; ++n)
                b[n] = *(const v2f*)&Bs[(k0 + kh) >> 1][(wn * 64 + n * 16 + L2) * 2];
#pragma unroll
            for (int s = 0; s < 2; ++s)
#pragma unroll
                for (int n = 0; n < 4; ++n)
                    acc[s * 4 + n] = __builtin_amdgcn_wmma_f32_16x16x4_f32(
                        false, a[s], false, b[n], (short)0, acc[s * 4 + n], false, false);
        }
        __syncthreads();
    }

    // ---- epilogue: C layout = VGPR r -> row (+8 for hi lane half), lane L2 -> col
#pragma unroll
    for (int s = 0; s < 2; ++s) {
        const int row_base = m0 + wm * 32 + s * 16 + (lane >> 4) * 8;
#pragma unroll
        for (int n = 0; n < 4; ++n) {
            int col = n0 + wn * 64 + n * 16 + L2;
            if (col < N) {
                float bb = bias ? bias[col] : 0.0f;
#pragma unroll
                for (int r = 0; r < 8; ++r) {
                    float v = acc[s * 4 + n][r] + bb;
                    if (act == 1)       v = tanhf(v);
                    else if (act == 2)  v = 1.0f / (1.0f + __expf(-v));
                    else if (act == 3)  v = DECAY_SCALE / (1.0f + __expf(-v));
                    Y[(size_t)(row_base + r) * N + col] = v;
                }
            }
        }
    }
}

// kk = L2-normalize(k*k_k) per head; k' = k*(1+(a-1)*k_a); a_vec=-kk; b_vec=kk*a
__global__ __launch_bounds__(64) void post_kv_kernel(
    const float* __restrict__ k, const float* __restrict__ a,
    const float* __restrict__ k_k, const float* __restrict__ k_a,
    float* __restrict__ kout, float* __restrict__ avec, float* __restrict__ bvec)
{
    int mh = blockIdx.x;
    int m = mh / NHEADS, h = mh % NHEADS;
    int vd = threadIdx.x;
    int d  = h * HEADD + vd;
    size_t idx = (size_t)m * D_MODEL + d;

    float kv = k[idx];
    float kk = kv * k_k[d];
    __shared__ float red[64];
    red[vd] = kk * kk;
    __syncthreads();
    for (int s = 32; s > 0; s >>= 1) { if (vd < s) red[vd] += red[vd + s]; __syncthreads(); }
    float kkn = kk / fmaxf(sqrtf(red[0]), 1e-12f);
    float av = a[idx];
    kout[idx] = kv * (1.0f + (av - 1.0f) * k_a[d]);
    avec[idx] = -kkn;
    bvec[idx] = kkn * av;    // may alias a[] (read-before-write, same idx)
}

// Sequential delta-rule scan. One block per (b,h); lane v owns state row S[v][*].
// Global loads for step t+1 are issued before step t's compute (double buffer).
__global__ __launch_bounds__(64) void rwkv7_scan(
    const float* __restrict__ r, const float* __restrict__ w,
    const float* __restrict__ k, const float* __restrict__ v,
    const float* __restrict__ avec, const float* __restrict__ bvec,
    float* __restrict__ o)
{
    int bh = blockIdx.x;
    int b = bh / NHEADS, h = bh % NHEADS;
    int vd = threadIdx.x;

    float S[HEADD];
#pragma unroll
    for (int j = 0; j < HEADD; ++j) S[j] = 0.0f;

    __shared__ float rs[64], we[64], ks[64], as_[64], bs[64];
    size_t base = ((size_t)b * TLEN) * D_MODEL + (size_t)h * HEADD + vd;

    // preload t = 0
    float rv = r[base], wv = w[base], kv = k[base];
    float av = avec[base], bv = bvec[base], vt = v[base];

    for (int t = 0; t < TLEN; ++t) {
        size_t idx = base + (size_t)t * D_MODEL;
        rs[vd]  = rv;
        we[vd]  = __expf(wv);
        ks[vd]  = kv;
        as_[vd] = av;
        bs[vd]  = bv;
        float vcur = vt;
        __syncthreads();

        // issue next step's loads; consumed only at the next LDS store
        if (t + 1 < TLEN) {
            size_t idx2 = idx + D_MODEL;
            rv = r[idx2]; wv = w[idx2]; kv = k[idx2];
            av = avec[idx2]; bv = bvec[idx2]; vt = v[idx2];
        }

        float sa = 0.0f;
#pragma unroll
        for (int j = 0; j < HEADD; ++j) sa = fmaf(S[j], as_[j], sa);

        float ot = 0.0f;
#pragma unroll
        for (int j = 0; j < HEADD; ++j) {
            float s = fmaf(S[j], we[j], fmaf(sa, bs[j], vcur * ks[j]));
            S[j] = s;
            ot = fmaf(s, rs[j], ot);
        }
        o[idx] = ot;
        __syncthreads();
    }
}

// GroupNorm per head + gate_output_correction. grid = M*H blocks of 64.
__global__ __launch_bounds__(64) void post_o_kernel(
    const float* __restrict__ o, const float* __restrict__ r,
    const float* __restrict__ k, const float* __restrict__ v,
    const float* __restrict__ g, const float* __restrict__ r_k,
    const float* __restrict__ gnw, const float* __restrict__ gnb,
    float* __restrict__ y)
{
    int mh = blockIdx.x;
    int m = mh / NHEADS, h = mh % NHEADS;
    int vd = threadIdx.x;
    int d  = h * HEADD + vd;
    size_t idx = (size_t)m * D_MODEL + d;

    float ov = o[idx];
    __shared__ float red[64];

    red[vd] = ov;
    __syncthreads();
    for (int s = 32; s > 0; s >>= 1) { if (vd < s) red[vd] += red[vd + s]; __syncthreads(); }
    float mu = red[0] * (1.0f / 64.0f);
    __syncthreads();

    float dv = ov - mu;
    red[vd] = dv * dv;
    __syncthreads();
    for (int s = 32; s > 0; s >>= 1) { if (vd < s) red[vd] += red[vd + s]; __syncthreads(); }
    float var = red[0] * (1.0f / 64.0f);
    __syncthreads();

    red[vd] = r[idx] * k[idx] * r_k[d];
    __syncthreads();
    for (int s = 32; s > 0; s >>= 1) { if (vd < s) red[vd] += red[vd + s]; __syncthreads(); }
    float corr = red[0];

    float on = dv * rsqrtf(var + GN_EPS) * gnw[d] + gnb[d];
    y[idx] = (on + corr * v[idx]) * g[idx];
}

extern "C" void kernel_launch(void* const* d_in, const int* in_sizes, int n_in,
                              void* d_out, int out_size, void* d_ws, size_t ws_size,
                              hipStream_t stream) {
    const float* X      = (const float*)d_in[0];
    const float* mix_r  = (const float*)d_in[1];
    const float* mix_w  = (const float*)d_in[2];
    const float* mix_k  = (const float*)d_in[3];
    const float* mix_v  = (const float*)d_in[4];
    const float* mix_a  = (const float*)d_in[5];
    const float* mix_g  = (const float*)d_in[6];
    const float* k_k    = (const float*)d_in[7];
    const float* k_a    = (const float*)d_in[8];
    const float* r_k    = (const float*)d_in[9];
    const float* Wr     = (const float*)d_in[10];
    const float* Wk     = (const float*)d_in[11];
    const float* Wv     = (const float*)d_in[12];
    const float* Wo     = (const float*)d_in[13];
    const float* wA     = (const float*)d_in[14];
    const float* wB     = (const float*)d_in[15];
    const float* w_bias = (const float*)d_in[16];
    const float* aA     = (const float*)d_in[17];
    const float* aB     = (const float*)d_in[18];
    const float* a_bias = (const float*)d_in[19];
    const float* gA     = (const float*)d_in[20];
    const float* gB     = (const float*)d_in[21];
    const float* gnw    = (const float*)d_in[22];
    const float* gnb    = (const float*)d_in[23];

    const int M = in_sizes[0] / D_MODEL;          // B*T = 8192
    const size_t MD = (size_t)M * D_MODEL;

    float* ws   = (float*)d_ws;
    float* r_b  = ws + 0 * MD;
    float* k_b  = ws + 1 * MD;
    float* v_b  = ws + 2 * MD;
    float* w_b  = ws + 3 * MD;   // log-decay; reused as gated-output after scan
    float* a_b  = ws + 4 * MD;   // a (icl rate); overwritten in place as b_vec
    float* g_b  = ws + 5 * MD;
    float* av_b = ws + 6 * MD;   // a_vec = -kk
    float* o_b  = ws + 7 * MD;   // scan output
    float* hw   = ws + 8 * MD;                       // [M,64]  tanh(xw@wA)
    float* ha   = hw + (size_t)M * 64;               // [M,64]  xa@aA
    float* hg   = ha + (size_t)M * 64;               // [M,160] sigmoid(xg@gA)
    float* oc_b = w_b;                               // alias: gated output

    dim3 blk(256);
    dim3 gD((D_MODEL + BN - 1) / BN, M / BM);
    dim3 g64((64 + BN - 1) / BN, M / BM);
    dim3 g160((160 + BN - 1) / BN, M / BM);

    // projections (token-shift mix fused into A loads)
    wmma_gemm_f32<<<gD,   blk, 0, stream>>>(X, mix_r, Wr, nullptr, r_b, M, D_MODEL, D_MODEL, 0);
    wmma_gemm_f32<<<gD,   blk, 0, stream>>>(X, mix_k, Wk, nullptr, k_b, M, D_MODEL, D_MODEL, 0);
    wmma_gemm_f32<<<gD,   blk, 0, stream>>>(X, mix_v, Wv, nullptr, v_b, M, D_MODEL, D_MODEL, 0);
    // w_lora: tanh(xw@wA) @ wB + bias -> DECAY_SCALE*sigmoid
    wmma_gemm_f32<<<g64,  blk, 0, stream>>>(X, mix_w, wA, nullptr, hw, M, 64, D_MODEL, 1);
    wmma_gemm_f32<<<gD,   blk, 0, stream>>>(hw, nullptr, wB, w_bias, w_b, M, D_MODEL, 64, 3);
    // a_lora: (xa@aA) @ aB + bias -> sigmoid
    wmma_gemm_f32<<<g64,  blk, 0, stream>>>(X, mix_a, aA, nullptr, ha, M, 64, D_MODEL, 0);
    wmma_gemm_f32<<<gD,   blk, 0, stream>>>(ha, nullptr, aB, a_bias, a_b, M, D_MODEL, 64, 2);
    // g_lora: sigmoid(xg@gA) @ gB
    wmma_gemm_f32<<<g160, blk, 0, stream>>>(X, mix_g, gA, nullptr, hg, M, 160, D_MODEL, 2);
    wmma_gemm_f32<<<gD,   blk, 0, stream>>>(hg, nullptr, gB, nullptr, g_b, M, D_MODEL, 160, 0);

    // kk normalization, k update, removal/replacement directions
    post_kv_kernel<<<M * NHEADS, 64, 0, stream>>>(k_b, a_b, k_k, k_a, k_b, av_b, a_b);

    // sequential delta-rule recurrence over T, one block per (b,h)
    rwkv7_scan<<<(M / TLEN) * NHEADS, 64, 0, stream>>>(r_b, w_b, k_b, v_b, av_b, a_b, o_b);

    // GroupNorm + correction + gate
    post_o_kernel<<<M * NHEADS, 64, 0, stream>>>(o_b, r_b, k_b, v_b, g_b, r_k, gnw, gnb, oc_b);

    // output projection
    wmma_gemm_f32<<<gD, blk, 0, stream>>>(oc_b, nullptr, Wo, nullptr, (float*)d_out, M, D_MODEL, D_MODEL, 0);
}